// RegularInputConversion_34308198761122
// MI455X (gfx1250) — compile-verified
//
#include <hip/hip_runtime.h>

// MI455X / gfx1250, wave32. e3nn-style Linear:
//   out0[t,w]   = (s @ W0) / sqrt(3)                     (tiny kernel, NT stores)
//   out1[t,w,i] = sum_u v[t,u,i] * W1[u,w] / sqrt(512)   (3x [8192x512]x[512x512] fp32 GEMM, WMMA f32)
//   out[..., 2048:8192] = 0                              (201 MB of NT zero stores)
// GEMM kernel: double-buffered LDS, async global->LDS (ASYNCcnt) when available.

typedef float v2f __attribute__((ext_vector_type(2)));
typedef float v4f __attribute__((ext_vector_type(4)));
typedef float v8f __attribute__((ext_vector_type(8)));
typedef int   v4i __attribute__((ext_vector_type(4)));

#define T_TOKENS   8192
#define KDIM       512
#define WDIM       512
#define OUT_STRIDE 8192      // 512*16 floats per token
#define BM         32        // rows per block
#define BN         128       // cols per block
#define KC         32        // K-chunk
#define NCHUNK     (KDIM / KC)
#define A_STRIDE   100       // LDS row stride (floats); 96 used; 400B (16B-aligned), conflict-free frag reads
#define B_STRIDE   144       // LDS row stride (floats); 128 used; 576B (16B-aligned), half-wave bank split

#define INV_SQRT3   0.57735026918962576f
#define INV_SQRTMUL 0.044194173824159216f   // 1/sqrt(512)

// ---- CDNA5 async global->LDS path (guarded; falls back to VGPR copies) ----
// Probed prototype: (v4i addrspace(1)* gsrc, v4i addrspace(3)* ldst, int offset, int cpol)
#if defined(__AMDGCN__) && __has_builtin(__builtin_amdgcn_global_load_async_to_lds_b128)
#define USE_ASYNC_LDS 1
#define ASYNC_COPY16(ldsdst, glbsrc)                                                   \
    __builtin_amdgcn_global_load_async_to_lds_b128(                                    \
        (__attribute__((address_space(1))) v4i*)(void*)(glbsrc),                       \
        (__attribute__((address_space(3))) v4i*)(void*)(ldsdst), 0, 0)
#else
#define USE_ASYNC_LDS 0
#define ASYNC_COPY16(ldsdst, glbsrc) (*(v4f*)(ldsdst) = *(const v4f*)(glbsrc))
#endif

#if __has_builtin(__builtin_amdgcn_s_wait_asynccnt)
#define ASYNC_WAIT(n) __builtin_amdgcn_s_wait_asynccnt(n)
#elif defined(__AMDGCN__)
#define ASYNC_WAIT(n) asm volatile("s_wait_asynccnt %0" ::"i"(n) : "memory")
#else
#define ASYNC_WAIT(n)
#endif

// ---------------- out0 + zero-fill kernel ----------------
__global__ __launch_bounds__(256) void e3nn_out0_zero_kernel(
    const float* __restrict__ scalars,  // [T,3]
    const float* __restrict__ W0,       // [3,512]
    float* __restrict__ out)            // [T,8192]
{
    const int t   = blockIdx.x;
    const int tid = threadIdx.x;
    const float s0 = scalars[t * 3 + 0];
    const float s1 = scalars[t * 3 + 1];
    const float s2 = scalars[t * 3 + 2];
    float* outT = out + (size_t)t * OUT_STRIDE;

    #pragma unroll
    for (int w = tid; w < 512; w += 256) {
        float r = (s0 * W0[w] + s1 * W0[512 + w] + s2 * W0[1024 + w]) * INV_SQRT3;
        __builtin_nontemporal_store(r, outT + w);
    }

    // zeros: [2048, 8192) -> 1536 x v4f, non-temporal (don't pollute L2)
    v4f z = {};
    v4f* zp = (v4f*)(outT + 2048);
    #pragma unroll
    for (int i = tid; i < 1536; i += 256)
        __builtin_nontemporal_store(z, zp + i);
}

// ---------------- WMMA fp32 GEMM kernel (out1) ----------------
// grid = (WDIM/BN, T/BM) ; block = 256 (8 waves)
// Wave wv: rowTile rt = wv&1, colTiles {2*(wv>>1), 2*(wv>>1)+1}, all 3 components.
__global__ __launch_bounds__(256) void e3nn_out1_wmma_kernel(
    const float* __restrict__ V,    // [T, 1536]  (u,i interleaved: v[t, 3u+i])
    const float* __restrict__ W1,   // [512, 512]
    float* __restrict__ out)        // [T, 8192]
{
    __shared__ __align__(16) float As[2][BM * A_STRIDE];  // raw interleaved rows: As[.][r][kk*3+comp]
    __shared__ __align__(16) float Bs[2][KC * B_STRIDE];  // Bs[.][kk][c]

    const int tid  = threadIdx.x;
    const int lane = tid & 31;
    const int wv   = tid >> 5;
    const int rt   = wv & 1;
    const int ct0  = (wv >> 1) * 2;
    const int n    = lane & 15;          // N index within 16x16 tile
    const int klo  = (lane >> 4) * 2;    // K pair: lanes 0-15 -> {0,1}, 16-31 -> {2,3}
    const int rowBase = blockIdx.y * BM;
    const int colBase = blockIdx.x * BN;

    v8f acc[3][2];
    {
        v8f zero = {};
        #pragma unroll
        for (int c = 0; c < 3; ++c)
            #pragma unroll
            for (int q = 0; q < 2; ++q)
                acc[c][q] = zero;
    }

    // staging thread mapping
    const int ar = tid >> 3;            // 0..31 : A row
    const int am = tid & 7;             // float4 slot within row (3 per thread)
    const int bk = tid >> 5;            // 0..7  : B row group
    const int bc = (tid & 31) * 4;      // 0..124: B col (float4)

    const float* aRow = V + (size_t)(rowBase + ar) * 1536;

    // per-thread share of one chunk: 3 A + 4 B = 7 async b128 ops
    auto stage = [&](int buf, int k0) {
        #pragma unroll
        for (int j = 0; j < 3; ++j) {
            const int f4 = am + 8 * j;                          // 0..23
            ASYNC_COPY16(&As[buf][ar * A_STRIDE + f4 * 4], aRow + k0 * 3 + f4 * 4);
        }
        #pragma unroll
        for (int q = 0; q < 4; ++q) {
            const int kk = bk + 8 * q;
            ASYNC_COPY16(&Bs[buf][kk * B_STRIDE + bc],
                         W1 + (size_t)(k0 + kk) * WDIM + colBase + bc);
        }
    };

    stage(0, 0);                                   // prologue: chunk 0 -> buf 0

    for (int i = 0; i < NCHUNK; ++i) {
        const int buf = i & 1;
        if (i + 1 < NCHUNK) {
            // issue next chunk into the other buffer (freed by the barrier that
            // ended iteration i-1), then wait for THIS chunk: in-order ASYNCcnt,
            // 7 outstanding == only chunk i+1's ops remain.
            stage(buf ^ 1, (i + 1) * KC);
#if USE_ASYNC_LDS
            ASYNC_WAIT(7);
#endif
        } else {
#if USE_ASYNC_LDS
            ASYNC_WAIT(0);
#endif
        }
        __syncthreads();                           // chunk i visible to all waves

        const float* asr = &As[buf][(rt * 16 + n) * A_STRIDE];
        const float* bsb = &Bs[buf][0];

        #pragma unroll
        for (int s = 0; s < KC / 4; ++s) {
            const int kc0 = 4 * s + klo;
            v2f a[3];
            #pragma unroll
            for (int c = 0; c < 3; ++c) {
                a[c].x = asr[kc0 * 3 + c];          // K = kc0
                a[c].y = asr[kc0 * 3 + 3 + c];      // K = kc0+1
            }
            v2f b[2];
            #pragma unroll
            for (int q = 0; q < 2; ++q) {
                const int off = kc0 * B_STRIDE + (ct0 + q) * 16 + n;
                b[q].x = bsb[off];
                b[q].y = bsb[off + B_STRIDE];
            }
            #pragma unroll
            for (int c = 0; c < 3; ++c)
                #pragma unroll
                for (int q = 0; q < 2; ++q)
                    acc[c][q] = __builtin_amdgcn_wmma_f32_16x16x4_f32(
                        /*neg_a=*/false, a[c], /*neg_b=*/false, b[q],
                        /*c_mod=*/(short)0, acc[c][q],
                        /*reuse_a=*/false, /*reuse_b=*/false);
        }
        __syncthreads();                           // release buf for chunk i+2
    }

    // --- epilogue: scale by 1/sqrt(512), scatter stride-3, non-temporal ---
    const int mh = (lane >> 4) * 8;     // VGPR r -> row M = r + mh
    #pragma unroll
    for (int c = 0; c < 3; ++c) {
        #pragma unroll
        for (int q = 0; q < 2; ++q) {
            const int col = colBase + (ct0 + q) * 16 + n;
            float* op = out + 512 + (size_t)col * 3 + c;
            #pragma unroll
            for (int r = 0; r < 8; ++r) {
                const int row = rowBase + rt * 16 + mh + r;
                __builtin_nontemporal_store(acc[c][q][r] * INV_SQRTMUL,
                                            op + (size_t)row * OUT_STRIDE);
            }
        }
    }
}

extern "C" void kernel_launch(void* const* d_in, const int* in_sizes, int n_in,
                              void* d_out, int out_size, void* d_ws, size_t ws_size,
                              hipStream_t stream) {
    const float* vectors = (const float*)d_in[0];   // [4,2048,1536]
    const float* scalars = (const float*)d_in[1];   // [4,2048,3]
    const float* W0      = (const float*)d_in[2];   // [3,512]
    const float* W1      = (const float*)d_in[3];   // [512,512]
    float* out = (float*)d_out;                     // [4,2048,8192]

    e3nn_out0_zero_kernel<<<dim3(T_TOKENS), dim3(256), 0, stream>>>(scalars, W0, out);
    e3nn_out1_wmma_kernel<<<dim3(WDIM / BN, T_TOKENS / BM), dim3(256), 0, stream>>>(vectors, W1, out);
}